// AdaptiveRiemannianParallelAttention_56049323213550
// MI455X (gfx1250) — compile-verified
//
#include <hip/hip_runtime.h>

#define BN 2048
#define C 64
#define D 128
#define H 8
#define TOTAL_CH 144
#define EPS 1e-5f

// padded LDS row strides (floats). 132 mod 64 = 4, 66 mod 64 = 2 -> conflict-free.
#define XS 132
#define AS 66

typedef float v2f __attribute__((ext_vector_type(2)));
typedef float v8f __attribute__((ext_vector_type(8)));

// ---------------------------------------------------------------------------
// Prep: mu_sub = 0.5 * (mu_log[ci][cj] + mu_log[cj][ci])  -- batch-independent
// ---------------------------------------------------------------------------
__global__ __launch_bounds__(C) void musub_prep(const float* __restrict__ mu_log,
                                                const int* __restrict__ cidx,
                                                float* __restrict__ musub) {
    int i = blockIdx.x;
    int j = threadIdx.x;
    int ci = cidx[i];
    int cj = cidx[j];
    musub[i * C + j] = 0.5f * (mu_log[ci * TOTAL_CH + cj] + mu_log[cj * TOTAL_CH + ci]);
}

// ---------------------------------------------------------------------------
// Main: per-batch  S = x x^T/D + eps I  (WMMA f32 16x16x4),
//       A = I + S, invert via Gauss-Jordan in LDS,
//       L = 2I - 4 A^{-1} - mu_sub,  bias[h] = L * head_scales[h]
// ---------------------------------------------------------------------------
__global__ __launch_bounds__(256) void riem_bias_kernel(
    const float* __restrict__ x,             // (BN, C, D)
    const float* __restrict__ head_scales,   // (H)
    const unsigned char* __restrict__ mask_space, // (BN, C) bool
    const float* __restrict__ musub,         // (C, C)
    float* __restrict__ bias_out,            // (BN, H, C, C)
    float* __restrict__ L_out)               // (BN, C, C)
{
    __shared__ float xs[C * XS];
    __shared__ float As[C * AS];
    __shared__ float mrow[C];

    const int tid = threadIdx.x;
    const int b = blockIdx.x;
    const float* xb = x + (size_t)b * (C * D);

    // per-head scales (wave-uniform -> scalar loads)
    float hs[H];
#pragma unroll
    for (int h = 0; h < H; ++h) hs[h] = head_scales[h];

    // mask flags -> LDS
    if (tid < C) mrow[tid] = mask_space[(size_t)b * C + tid] ? 1.f : 0.f;

    // stage x[b] into padded LDS rows, float4 coalesced
#pragma unroll
    for (int it = 0; it < 8; ++it) {
        int v = tid + it * 256;          // 2048 float4 chunks total
        int row = v >> 5;
        int c4 = (v & 31) << 2;
        float4 val = *(const float4*)(xb + row * D + c4);
        *(float4*)(&xs[row * XS + c4]) = val;
    }
    __syncthreads();

    // --- WMMA phase: 16 tiles of 16x16, 8 waves, 2 tiles per wave -----------
    const int lane = tid & 31;
    const int wv = tid >> 5;
    const int lhi = lane >> 4;   // 0: K0,K1   1: K2,K3  (32-bit A/B layout)
    const int llo = lane & 15;

#pragma unroll
    for (int tt = 0; tt < 2; ++tt) {
        int t = wv + tt * 8;
        int mi = t >> 2;
        int ni = t & 3;
        const float* arow = &xs[(mi * 16 + llo) * XS + lhi * 2];
        const float* brow = &xs[(ni * 16 + llo) * XS + lhi * 2];

        v8f acc = {};
#pragma unroll
        for (int kk = 0; kk < D; kk += 4) {
            v2f a  = *(const v2f*)(arow + kk);
            v2f bb = *(const v2f*)(brow + kk);
            acc = __builtin_amdgcn_wmma_f32_16x16x4_f32(
                false, a, false, bb, (short)0, acc, false, false);
        }

        // scatter D per 32-bit C/D layout: VGPR r -> row r (lanes 0-15) / r+8
#pragma unroll
        for (int r = 0; r < 8; ++r) {
            int row = mi * 16 + r + lhi * 8;
            int col = ni * 16 + llo;
            float diag = (row == col) ? 1.f : 0.f;
            float s = acc[r] * (1.f / (float)D) + EPS * diag;
            float m = fminf(mrow[row] + mrow[col], 1.f);
            s = s * (1.f - m) + diag * m;       // masked rows/cols -> identity
            As[row * AS + col] = s + diag;      // A = I + S
        }
    }
    __syncthreads();

    // --- In-place Gauss-Jordan inversion of SPD A (no pivoting) -------------
    // 4 threads per row, 16 columns each.
    const int gi = tid >> 2;
    const int j0 = (tid & 3) << 4;
    for (int k = 0; k < C; ++k) {
        float p = 1.f / As[k * AS + k];
        float aik = As[gi * AS + k];
        float rk[16];
#pragma unroll
        for (int j = 0; j < 16; ++j) rk[j] = As[k * AS + j0 + j];
        __syncthreads();
        if (gi == k) {
#pragma unroll
            for (int j = 0; j < 16; ++j) {
                int col = j0 + j;
                As[k * AS + col] = (col == k) ? p : rk[j] * p;
            }
        } else {
#pragma unroll
            for (int j = 0; j < 16; ++j) {
                int col = j0 + j;
                float rnew = (col == k) ? p : rk[j] * p;
                float aij = (col == k) ? 0.f : As[gi * AS + col];
                As[gi * AS + col] = aij - aik * rnew;
            }
        }
        __syncthreads();
    }

    // --- Fused epilogue: L = 2I - 4 Ainv - mu_sub ; bias[h] = L * hs[h] -----
    float* Lb = L_out + (size_t)b * (C * C);
    float* Bb = bias_out + (size_t)b * (H * C * C);
#pragma unroll
    for (int it = 0; it < 4; ++it) {
        int v = tid + it * 256;       // float4 index 0..1023
        int e = v << 2;
        int i = e >> 6;
        int j = e & 63;
        float4 mu4 = *(const float4*)(musub + e);
        float l0 = ((i == j + 0) ? 2.f : 0.f) - 4.f * As[i * AS + j + 0] - mu4.x;
        float l1 = ((i == j + 1) ? 2.f : 0.f) - 4.f * As[i * AS + j + 1] - mu4.y;
        float l2 = ((i == j + 2) ? 2.f : 0.f) - 4.f * As[i * AS + j + 2] - mu4.z;
        float l3 = ((i == j + 3) ? 2.f : 0.f) - 4.f * As[i * AS + j + 3] - mu4.w;
        *(float4*)(Lb + e) = make_float4(l0, l1, l2, l3);
#pragma unroll
        for (int h = 0; h < H; ++h) {
            *(float4*)(Bb + h * (C * C) + e) =
                make_float4(l0 * hs[h], l1 * hs[h], l2 * hs[h], l3 * hs[h]);
        }
    }
}

// ---------------------------------------------------------------------------
extern "C" void kernel_launch(void* const* d_in, const int* in_sizes, int n_in,
                              void* d_out, int out_size, void* d_ws, size_t ws_size,
                              hipStream_t stream) {
    const float* x            = (const float*)d_in[0];
    const float* head_scales  = (const float*)d_in[1];
    const float* mu_log       = (const float*)d_in[2];
    const int*   channel_idx  = (const int*)d_in[3];
    const unsigned char* mask = (const unsigned char*)d_in[4];

    float* out      = (float*)d_out;
    float* bias_out = out;                                   // (BN,H,C,C) first
    float* L_out    = out + (size_t)BN * H * C * C;          // then (BN,C,C)
    float* musub    = (float*)d_ws;                          // C*C floats

    musub_prep<<<C, C, 0, stream>>>(mu_log, channel_idx, musub);
    riem_bias_kernel<<<BN, 256, 0, stream>>>(x, head_scales, mask, musub,
                                             bias_out, L_out);
}